// SDF_84104049590471
// MI455X (gfx1250) — compile-verified
//
#include <hip/hip_runtime.h>
#include <math.h>

// Problem constants (from reference): B=1, V=6890, NF=13776, P=4096
#define NFACES 13776
#define NPTS   4096
#define NSLICE 7                    // face slices (861 tiles = 7 * 123)
#define NTILES (NFACES / 16)        // 861 16-face tiles, exact
#define TPS    (NTILES / NSLICE)    // 123 tiles per slice, exact
#define EPSF   1e-12f

typedef float v2f __attribute__((ext_vector_type(2)));
typedef float v8f __attribute__((ext_vector_type(8)));

// Raw-hardware helpers: avoid ocml range fix-ups (inputs are clamped/biased).
__device__ __forceinline__ float fsqrt_raw(float x) { return __builtin_amdgcn_sqrtf(x); }
__device__ __forceinline__ float frcp_raw(float x)  { return __builtin_amdgcn_rcpf(x); }

// Fast atan2: 5-coefficient odd minimax on [0,1] + quadrant fixup
// (~1e-5 rad error/term). The winding-number sign decision has ~pi of margin
// over 13776 summed terms, so this comfortably replaces ocml atan2f.
__device__ __forceinline__ float fast_atan2(float y, float x)
{
  const float ax = fabsf(x), ay = fabsf(y);
  const float mx = fmaxf(ax, ay), mn = fminf(ax, ay);
  const float a = mn * frcp_raw(mx + 1e-30f);    // in [0,1]
  const float s = a * a;
  float r = fmaf(s, 0.0208351f, -0.0851330f);
  r = fmaf(s, r, 0.1801410f);
  r = fmaf(s, r, -0.3302995f);
  r = fmaf(s, r, 0.9998660f);
  r = a * r;
  r = (ay > ax) ? (1.57079632679f - r) : r;
  r = (x < 0.f) ? (3.14159265359f - r) : r;
  return copysignf(r, y);
}

// ---------------------------------------------------------------------------
// Kernel 1: per-face precompute. Packs, for each face, the seven homogeneous
// K=4 B-matrix columns (so WMMA computes p.v + const directly) plus the
// per-face scalars needed by the closest-point epilogue.
// col[f][q][4]: q=0:(ab,-ab.a) 1:(ac,-ac.a) 2:(-2a,a.a) 3:(-Nw,D0)
//              4:(-(a+b),a.b) 5:(-(b+c),b.c) 6:(-(c+a),c.a)
// scal[f][8]:  ab2, ac2, ab.ac, 1/(bc2+eps), 1/(ab2+eps), 1/(ac2+eps), 0, 0
// ---------------------------------------------------------------------------
__global__ void sdf_precompute(const int* __restrict__ faces,
                               const float* __restrict__ verts,
                               float* __restrict__ col,
                               float* __restrict__ scal)
{
  int f = blockIdx.x * blockDim.x + threadIdx.x;
  if (f >= NFACES) return;
  int i0 = faces[3*f+0], i1 = faces[3*f+1], i2 = faces[3*f+2];
  float ax = verts[3*i0+0], ay = verts[3*i0+1], az = verts[3*i0+2];
  float bx = verts[3*i1+0], by = verts[3*i1+1], bz = verts[3*i1+2];
  float cx = verts[3*i2+0], cy = verts[3*i2+1], cz = verts[3*i2+2];

  float abx=bx-ax, aby=by-ay, abz=bz-az;
  float acx=cx-ax, acy=cy-ay, acz=cz-az;
  float cbx=cx-bx, cby=cy-by, cbz=cz-bz;
  float ab2  = abx*abx+aby*aby+abz*abz;
  float ac2  = acx*acx+acy*acy+acz*acz;
  float bc2  = cbx*cbx+cby*cby+cbz*cbz;
  float abac = abx*acx+aby*acy+abz*acz;
  float aa   = ax*ax+ay*ay+az*az;
  float ab_a = abx*ax+aby*ay+abz*az;
  float ac_a = acx*ax+acy*ay+acz*az;
  // cross products
  float bxcx = by*cz-bz*cy, bxcy = bz*cx-bx*cz, bxcz = bx*cy-by*cx;
  float cxax = cy*az-cz*ay, cxay = cz*ax-cx*az, cxaz = cx*ay-cy*ax;
  float axbx = ay*bz-az*by, axby = az*bx-ax*bz, axbz = ax*by-ay*bx;
  float D0  = ax*bxcx + ay*bxcy + az*bxcz;
  float nwx = bxcx+cxax+axbx, nwy = bxcy+cxay+axby, nwz = bxcz+cxaz+axbz;
  float a_b = ax*bx+ay*by+az*bz;
  float b_c = bx*cx+by*cy+bz*cz;
  float c_a = cx*ax+cy*ay+cz*az;

  float* cf = col + (size_t)f * 28;
  cf[ 0]=abx;       cf[ 1]=aby;       cf[ 2]=abz;       cf[ 3]=-ab_a;
  cf[ 4]=acx;       cf[ 5]=acy;       cf[ 6]=acz;       cf[ 7]=-ac_a;
  cf[ 8]=-2.f*ax;   cf[ 9]=-2.f*ay;   cf[10]=-2.f*az;   cf[11]=aa;
  cf[12]=-nwx;      cf[13]=-nwy;      cf[14]=-nwz;      cf[15]=D0;
  cf[16]=-(ax+bx);  cf[17]=-(ay+by);  cf[18]=-(az+bz);  cf[19]=a_b;
  cf[20]=-(bx+cx);  cf[21]=-(by+cy);  cf[22]=-(bz+cz);  cf[23]=b_c;
  cf[24]=-(cx+ax);  cf[25]=-(cy+ay);  cf[26]=-(cz+az);  cf[27]=c_a;

  float* sf = scal + (size_t)f * 8;
  sf[0]=ab2;  sf[1]=ac2;  sf[2]=abac; sf[3]=1.f/(bc2+EPSF);
  sf[4]=1.f/(ab2+EPSF);   sf[5]=1.f/(ac2+EPSF); sf[6]=0.f; sf[7]=0.f;
}

// ---------------------------------------------------------------------------
// Kernel 2: WMMA tile kernel. Each wave owns a 16-point tile; for each
// 16-face tile it issues 7x V_WMMA_F32_16X16X4_F32 producing all affine
// per-pair quantities lane-locally, then runs the branchless Ericson
// closest-point + winding-number epilogue on 8 (point,face) pairs per lane.
// A layout (f32 16x4): lane<16 supplies K=0,1 (x,y); lane>=16 K=2,3 (z,1).
// B layout (f32 4x16): lane<16 supplies K=0,1; lane>=16 K=2,3 of column l%16.
// D layout: VGPR r, lane l -> point M = r + 8*(l/16), face N = l%16.
// ---------------------------------------------------------------------------
__global__ __launch_bounds__(256)
void sdf_tiles(const float* __restrict__ pts,
               const float* __restrict__ col,
               const float* __restrict__ scal,
               float* __restrict__ pmin,
               float* __restrict__ pwind)
{
  const int lane = threadIdx.x & 31;
  const int wave = threadIdx.x >> 5;
  const int h    = lane >> 4;        // lane half (K-split / row-block)
  const int n16  = lane & 15;        // face column within tile
  const int pointBase = (blockIdx.x * 8 + wave) * 16;
  const int slice = blockIdx.y;

  // A matrix: homogeneous point rows (x,y,z,1), row M = n16.
  v2f Av;
  {
    const float* p = pts + 3 * (pointBase + n16);
    float px = p[0], py = p[1], pz = p[2];
    Av.x = h ? pz   : px;
    Av.y = h ? 1.0f : py;
  }

  // |p|^2 for the 8 D-rows this lane owns (point = pointBase + 8*h + r).
  float pp[8];
  #pragma unroll
  for (int r = 0; r < 8; ++r) {
    const float* p = pts + 3 * (pointBase + 8*h + r);
    pp[r] = p[0]*p[0] + p[1]*p[1] + p[2]*p[2];
  }

  float minsq[8], wind[8];
  #pragma unroll
  for (int r = 0; r < 8; ++r) { minsq[r] = 3.4e38f; wind[r] = 0.f; }

  const int t0 = slice * TPS, t1 = t0 + TPS;
  for (int t = t0; t < t1; ++t) {
    const int f = t * 16 + n16;
    const float* cf = col + (size_t)f * 28 + 2*h;   // 8B-aligned
    v2f Bq[7];
    #pragma unroll
    for (int q = 0; q < 7; ++q)
      Bq[q] = *(const v2f*)(cf + 4*q);
    __builtin_prefetch(cf + 16*28, 0, 1);           // next tile's columns

    const float4* sp = (const float4*)(scal + (size_t)f * 8);
    const float4 s0 = sp[0], s1 = sp[1];
    const float ab2 = s0.x, ac2 = s0.y, abac = s0.z, rbc2 = s0.w;
    const float rab2 = s1.x, rac2 = s1.y;

    const v8f cz = {0.f,0.f,0.f,0.f,0.f,0.f,0.f,0.f};
    v8f Dv[7];
    #pragma unroll
    for (int q = 0; q < 7; ++q)
      Dv[q] = __builtin_amdgcn_wmma_f32_16x16x4_f32(
                  false, Av, false, Bq[q], (short)0, cz, false, false);

    #pragma unroll
    for (int r = 0; r < 8; ++r) {
      const float ppr = pp[r];
      const float d1  = Dv[0][r];
      const float d2  = Dv[1][r];
      const float pa2 = fmaxf(ppr + Dv[2][r], 0.f);
      const float d3 = d1 - ab2,  d4 = d2 - abac;
      const float d5 = d1 - abac, d6 = d2 - ac2;
      const float sq_b = pa2 - 2.f*d1 + ab2;
      const float sq_c = pa2 - 2.f*d2 + ac2;
      const float va = d3*d6 - d5*d4;
      const float vb = d5*d2 - d1*d6;
      const float vc = d1*d4 - d3*d2;
      const float sq_ab = pa2 - d1*d1*rab2;
      const float sq_ac = pa2 - d2*d2*rac2;
      const float e = d4 - d3;
      const float sq_bc = sq_b - e*e*rbc2;
      const float rden = frcp_raw(va + vb + vc + EPSF);
      const float v = vb * rden, w = vc * rden;
      const float sq_in = pa2 - 2.f*(v*d1 + w*d2)
                        + v*v*ab2 + 2.f*v*w*abac + w*w*ac2;
      // selection priority matches nested jnp.where (innermost applied first)
      float sq = ((va<=0.f) && (e>=0.f) && ((d5-d6)>=0.f)) ? sq_bc : sq_in;
      sq = ((vb<=0.f) && (d2>=0.f) && (d6<=0.f)) ? sq_ac : sq;
      sq = ((vc<=0.f) && (d1>=0.f) && (d3<=0.f)) ? sq_ab : sq;
      sq = ((d6>=0.f) && (d5<=d6))               ? sq_c  : sq;
      sq = ((d3>=0.f) && (d4<=d3))               ? sq_b  : sq;
      sq = ((d1<=0.f) && (d2<=0.f))              ? pa2   : sq;
      minsq[r] = fminf(minsq[r], sq);

      // generalized winding number term (raw v_sqrt: args are >= EPS)
      const float la = fsqrt_raw(pa2 + EPSF);
      const float lb = fsqrt_raw(fmaxf(sq_b, 0.f) + EPSF);
      const float lc = fsqrt_raw(fmaxf(sq_c, 0.f) + EPSF);
      const float rarb = ppr + Dv[4][r];
      const float rbrc = ppr + Dv[5][r];
      const float rcra = ppr + Dv[6][r];
      const float den = la*lb*lc + rarb*lc + rbrc*la + rcra*lb;
      wind[r] += fast_atan2(Dv[3][r], den);
    }
  }

  // Reduce across the 16 face-columns (lanes sharing l/16): min + sum.
  #pragma unroll
  for (int r = 0; r < 8; ++r) {
    float mn = minsq[r], sm = wind[r];
    #pragma unroll
    for (int msk = 1; msk < 16; msk <<= 1) {
      mn = fminf(mn, __shfl_xor(mn, msk, 16));
      sm += __shfl_xor(sm, msk, 16);
    }
    if (n16 == 0) {
      const int m = pointBase + 8*h + r;
      pmin [slice * NPTS + m] = mn;
      pwind[slice * NPTS + m] = sm;
    }
  }
}

// ---------------------------------------------------------------------------
// Kernel 3: combine slice partials, final sign * distance.
// ---------------------------------------------------------------------------
__global__ void sdf_finalize(const float* __restrict__ pmin,
                             const float* __restrict__ pwind,
                             float* __restrict__ out)
{
  int m = blockIdx.x * blockDim.x + threadIdx.x;
  if (m >= NPTS) return;
  float mn = 3.4e38f, sm = 0.f;
  #pragma unroll
  for (int s = 0; s < NSLICE; ++s) {
    mn = fminf(mn, pmin[s * NPTS + m]);
    sm += pwind[s * NPTS + m];
  }
  const float dist = fsqrt_raw(fmaxf(mn, 0.f) + EPSF);
  const float wnum = sm * 0.15915494309189535f;   // / (2*pi)
  out[m] = (fabsf(wnum) >= 0.5f) ? dist : -dist;
}

// ---------------------------------------------------------------------------
extern "C" void kernel_launch(void* const* d_in, const int* in_sizes, int n_in,
                              void* d_out, int out_size, void* d_ws, size_t ws_size,
                              hipStream_t stream)
{
  const int*   faces = (const int*)  d_in[0];   // (NF,3) int32
  const float* verts = (const float*)d_in[1];   // (1,V,3) f32
  const float* pts   = (const float*)d_in[2];   // (1,P,3) f32
  float* out = (float*)d_out;                   // (1,P) f32
  float* ws  = (float*)d_ws;

  float* col   = ws;                                  // NF*28 floats
  float* scal  = ws + (size_t)NFACES * 28;            // NF*8 floats
  float* pmin  = ws + (size_t)NFACES * 36;            // NSLICE*P floats
  float* pwind = pmin + (size_t)NSLICE * NPTS;        // NSLICE*P floats

  sdf_precompute<<<(NFACES + 255) / 256, 256, 0, stream>>>(faces, verts, col, scal);
  sdf_tiles<<<dim3(NPTS / 128, NSLICE), 256, 0, stream>>>(pts, col, scal, pmin, pwind);
  sdf_finalize<<<(NPTS + 255) / 256, 256, 0, stream>>>(pmin, pwind, out);
}